// CatmullRomSplines_14894946583250
// MI455X (gfx1250) — compile-verified
//
#include <hip/hip_runtime.h>
#include <cmath>

// Catmull-Rom basis matrix (row-major), zero-padded to 4x16 as WMMA B operand.
__device__ __constant__ float c_A[4][4] = {
    {-0.5f,  1.5f, -1.5f,  0.5f},
    { 1.0f, -2.5f,  2.0f, -0.5f},
    {-0.5f,  0.0f,  0.5f,  0.0f},
    { 0.0f,  1.0f,  0.0f,  0.0f}};

typedef float v2f __attribute__((ext_vector_type(2)));
typedef float v8f __attribute__((ext_vector_type(8)));

#define NBLOCKS 1024
#define NTHREADS 256
#define WAVES_PER_BLOCK (NTHREADS / 32)

__global__ void __launch_bounds__(NTHREADS)
spline_sse_kernel(const float* __restrict__ ch1,
                  const float* __restrict__ CP,     // G x G x 2
                  const int*   __restrict__ idx,    // N x 2
                  const float* __restrict__ r,      // N x 2
                  float* __restrict__ partial,      // NBLOCKS
                  int N, int G)
{
    __shared__ float sW[WAVES_PER_BLOCK][32][8];   // per-wave W staging: 32 pts x (4 wx + 4 wy)
    __shared__ float sRed[WAVES_PER_BLOCK];

    const int lane   = threadIdx.x & 31;
    const int w      = threadIdx.x >> 5;
    const int n      = lane & 15;          // WMMA column / point-within-half
    const int hiHalf = lane >> 4;          // 0: K=0,1  1: K=2,3 (A layout halves)
    const int kbase  = hiHalf ? 2 : 0;

    // Loop-invariant B fragment: B[k][n] = A[k][n] for n<4, else 0 (4x16 zero-padded).
    // 32-bit B layout mirrors A: lanes 0-15 hold K=0 (v.x), K=1 (v.y); lanes 16-31 K=2,3.
    v2f bfrag;
    bfrag.x = (n < 4) ? c_A[kbase][n]     : 0.0f;
    bfrag.y = (n < 4) ? c_A[kbase + 1][n] : 0.0f;

    const int waveId = (blockIdx.x * blockDim.x + threadIdx.x) >> 5;
    const int nWaves = (gridDim.x * blockDim.x) >> 5;

    float acc = 0.0f;

    for (int base = waveId * 32; base < N; base += nWaves * 32) {
        const int  p     = base + lane;
        const bool valid = (p < N);

        float xf = 0.0f, yf = 0.0f, c0 = 0.0f, c1 = 0.0f;
        int   i0 = 1, i1 = 1;
        if (valid) {
            float2 rv = ((const float2*)r)[p];
            xf = rv.x - floorf(rv.x);
            yf = rv.y - floorf(rv.y);
            float2 cv = ((const float2*)ch1)[p];
            c0 = cv.x; c1 = cv.y;
            int2 iv = ((const int2*)idx)[p];
            i0 = iv.x; i1 = iv.y;
        }

        // ---- WMMA basis evaluation: W = P(t) @ A, 16 points per instruction ----
        // x of point (lane&15) for the low half-tile, of point 16+(lane&15) for high.
        const float xs_lo = __shfl(xf, n, 32);
        const float xs_hi = __shfl(xf, 16 + n, 32);
        const float ys_lo = __shfl(yf, n, 32);
        const float ys_hi = __shfl(yf, 16 + n, 32);

        // A-matrix 16x4 f32 layout: lanes 0-15 -> {K0=t^3, K1=t^2}; lanes 16-31 -> {K2=t, K3=1}
        v2f ax_lo, ax_hi, ay_lo, ay_hi;
        if (hiHalf == 0) {
            ax_lo.x = xs_lo * xs_lo * xs_lo; ax_lo.y = xs_lo * xs_lo;
            ax_hi.x = xs_hi * xs_hi * xs_hi; ax_hi.y = xs_hi * xs_hi;
            ay_lo.x = ys_lo * ys_lo * ys_lo; ay_lo.y = ys_lo * ys_lo;
            ay_hi.x = ys_hi * ys_hi * ys_hi; ay_hi.y = ys_hi * ys_hi;
        } else {
            ax_lo.x = xs_lo; ax_lo.y = 1.0f;
            ax_hi.x = xs_hi; ax_hi.y = 1.0f;
            ay_lo.x = ys_lo; ay_lo.y = 1.0f;
            ay_hi.x = ys_hi; ay_hi.y = 1.0f;
        }

        v8f cz = {0.0f, 0.0f, 0.0f, 0.0f, 0.0f, 0.0f, 0.0f, 0.0f};
        v8f dx_lo = __builtin_amdgcn_wmma_f32_16x16x4_f32(false, ax_lo, false, bfrag, (short)0, cz, false, false);
        v8f dx_hi = __builtin_amdgcn_wmma_f32_16x16x4_f32(false, ax_hi, false, bfrag, (short)0, cz, false, false);
        v8f dy_lo = __builtin_amdgcn_wmma_f32_16x16x4_f32(false, ay_lo, false, bfrag, (short)0, cz, false, false);
        v8f dy_hi = __builtin_amdgcn_wmma_f32_16x16x4_f32(false, ay_hi, false, bfrag, (short)0, cz, false, false);

        // ---- unstripe D (16x16 f32 C/D layout) through per-wave LDS ----
        // VGPR v, lanes 0-15: row M=v, col N=lane; lanes 16-31: row M=v+8, col N=lane-16.
        if (n < 4) {
#pragma unroll
            for (int v = 0; v < 8; ++v) {
                const int m = v + hiHalf * 8;
                sW[w][m][n]          = dx_lo[v];
                sW[w][m + 16][n]     = dx_hi[v];
                sW[w][m][n + 4]      = dy_lo[v];
                sW[w][m + 16][n + 4] = dy_hi[v];
            }
        }
        asm volatile("s_wait_dscnt 0" ::: "memory");  // wave-private region: wave-internal ordering

        float wx[4], wy[4];
#pragma unroll
        for (int k = 0; k < 4; ++k) {
            wx[k] = sW[w][lane][k];
            wy[k] = sW[w][lane][4 + k];
        }

        // ---- 4x4x2 gather (L2-resident control grid) + bicubic blend ----
        if (valid) {
            float s0 = 0.0f, s1 = 0.0f;
#pragma unroll
            for (int i = 0; i < 4; ++i) {
                const float* rowp = CP + ((size_t)(i0 - 1 + i) * G + (size_t)(i1 - 1)) * 2;
                float t0 = 0.0f, t1 = 0.0f;
#pragma unroll
                for (int j = 0; j < 4; ++j) {
                    float2 q = ((const float2*)rowp)[j];
                    t0 = fmaf(wy[j], q.x, t0);
                    t1 = fmaf(wy[j], q.y, t1);
                }
                s0 = fmaf(wx[i], t0, s0);
                s1 = fmaf(wx[i], t1, s1);
            }
            const float d0 = c0 - s0;
            const float d1 = c1 - s1;
            acc = fmaf(d0, d0, fmaf(d1, d1, acc));
        }
    }

    // ---- deterministic block reduction ----
#pragma unroll
    for (int off = 16; off > 0; off >>= 1) acc += __shfl_down(acc, off, 32);
    if (lane == 0) sRed[w] = acc;
    __syncthreads();
    if (w == 0) {
        float v = (lane < WAVES_PER_BLOCK) ? sRed[lane] : 0.0f;
#pragma unroll
        for (int off = 4; off > 0; off >>= 1) v += __shfl_down(v, off, 32);
        if (lane == 0) partial[blockIdx.x] = v;
    }
}

__global__ void __launch_bounds__(NTHREADS)
final_reduce_kernel(const float* __restrict__ partial, float* __restrict__ out, int n)
{
    __shared__ float sRed[WAVES_PER_BLOCK];
    const int lane = threadIdx.x & 31;
    const int w    = threadIdx.x >> 5;

    float acc = 0.0f;
    for (int i = threadIdx.x; i < n; i += blockDim.x) acc += partial[i];
#pragma unroll
    for (int off = 16; off > 0; off >>= 1) acc += __shfl_down(acc, off, 32);
    if (lane == 0) sRed[w] = acc;
    __syncthreads();
    if (w == 0) {
        float v = (lane < WAVES_PER_BLOCK) ? sRed[lane] : 0.0f;
#pragma unroll
        for (int off = 4; off > 0; off >>= 1) v += __shfl_down(v, off, 32);
        if (lane == 0) out[0] = v;
    }
}

extern "C" void kernel_launch(void* const* d_in, const int* in_sizes, int n_in,
                              void* d_out, int out_size, void* d_ws, size_t ws_size,
                              hipStream_t stream)
{
    const float* ch1     = (const float*)d_in[0];   // N x 2
    const float* CP_locs = (const float*)d_in[1];   // G x G x 2
    const int*   CP_idx  = (const int*)  d_in[2];   // N x 2
    const float* r       = (const float*)d_in[3];   // N x 2

    const int N  = in_sizes[0] / 2;
    const int gg = in_sizes[1] / 2;
    const int G  = (int)(sqrt((double)gg) + 0.5);

    float* partial = (float*)d_ws;                  // NBLOCKS floats of scratch
    float* out     = (float*)d_out;

    spline_sse_kernel<<<NBLOCKS, NTHREADS, 0, stream>>>(ch1, CP_locs, CP_idx, r, partial, N, G);
    final_reduce_kernel<<<1, NTHREADS, 0, stream>>>(partial, out, NBLOCKS);
}